// SSMLayer_91199335563472
// MI455X (gfx1250) — compile-verified
//
#include <hip/hip_runtime.h>
#include <hip/hip_bf16.h>

// SSM (S4-style) layer for MI455X / gfx1250.
// H=256 channels, N=64 state dim, L=4096 sequence length.
//
// Kernel 1 (1 workgroup of 256 threads = 8 wave32 per channel):
//   - build M = I - (s/2)A, P = I + (s/2)A in LDS
//   - Gauss-Jordan invert M  -> BL
//   - Ab = BL @ P  (WMMA f32 16x16x4 GEMM), Bb = s * BL @ B
//   - X = [Bb, Ab Bb, ..., Ab^63 Bb]  (63 matvecs, shuffle-reduced)
//   - MT = Ab^64 via 6 WMMA squarings
//   - 64 blocks: K-row = C @ X ; X <- MT @ X (WMMA GEMM)
// Kernel 2 (1 workgroup per channel): direct causal Toeplitz conv in LDS
//   (K and u both fit in LDS; FFT unnecessary at this FLOP count),
//   y = K (*) u + D*u.

#define SSM_H 256
#define SSM_N 64
#define SSM_L 4096
#define LDS_LD 65   // row stride (floats) for 64x64 LDS tiles, bank-conflict free

typedef __attribute__((ext_vector_type(2))) float v2f;
typedef __attribute__((ext_vector_type(8))) float v8f;

// D(64x64) = A(64x64) @ B(64x64); all operands in LDS with row stride LDS_LD.
// 8 waves: wave w computes output tiles 2w and 2w+1 (tile grid 4x4 of 16x16).
// Uses V_WMMA_F32_16X16X4_F32 (CDNA5), accumulating over 16 K-blocks of 4.
__device__ __forceinline__ void gemm64(const float* Am, const float* Bm, float* Dm) {
    __syncthreads();
    const int lane = threadIdx.x & 31;
    const int wave = threadIdx.x >> 5;
    const int l15  = lane & 15;
    const int lh   = lane >> 4;   // lane half: 0 or 1
#pragma unroll
    for (int tt = 0; tt < 2; ++tt) {
        const int t  = wave * 2 + tt;
        const int ti = (t >> 2) << 4;   // output row base
        const int tj = (t & 3) << 4;    // output col base
        v8f acc = {0.f, 0.f, 0.f, 0.f, 0.f, 0.f, 0.f, 0.f};
#pragma unroll
        for (int kb = 0; kb < 16; ++kb) {
            const int ks = kb * 4 + lh * 2;   // this lane-half's K offsets: ks, ks+1
            v2f a, b;
            // A 16x4 fragment: lane holds row (ti+l15), K = ks, ks+1
            a.x = Am[(ti + l15) * LDS_LD + ks];
            a.y = Am[(ti + l15) * LDS_LD + ks + 1];
            // B 4x16 fragment: VGPR0 = K-row ks, VGPR1 = K-row ks+1, col = tj+l15
            b.x = Bm[(ks)     * LDS_LD + tj + l15];
            b.y = Bm[(ks + 1) * LDS_LD + tj + l15];
            acc = __builtin_amdgcn_wmma_f32_16x16x4_f32(
                false, a, false, b, (short)0, acc, false, false);
        }
        // C/D layout: VGPR v -> row ti + v + 8*lane_half, col tj + l15
#pragma unroll
        for (int v = 0; v < 8; ++v)
            Dm[(ti + v + lh * 8) * LDS_LD + tj + l15] = acc[v];
    }
    __syncthreads();
}

__global__ __launch_bounds__(256)
void ssm_kernel_K(const float* __restrict__ Ain,      // (H,N,N)
                  const float* __restrict__ Bin,      // (H,N)
                  const float* __restrict__ Cin,      // (H,N)
                  const float* __restrict__ logstep,  // (H)
                  float* __restrict__ Kbuf) {         // (H,L) out
    const int h   = blockIdx.x;
    const int tid = threadIdx.x;

    __shared__ float sA[SSM_N][LDS_LD];
    __shared__ float sB[SSM_N][LDS_LD];
    __shared__ float sC[SSM_N][LDS_LD];
    __shared__ float fac[SSM_N], Cs[SSM_N], Bbs[SSM_N], Bsh[SSM_N];

    const float stp = expf(logstep[h]);
    const float hs  = 0.5f * stp;

    // Load: sA = I - hs*A, sC = I + hs*A, sB = I
    for (int e = tid; e < SSM_N * SSM_N; e += 256) {
        const int i = e >> 6, j = e & 63;
        const float a  = Ain[h * SSM_N * SSM_N + e];
        const float id = (i == j) ? 1.f : 0.f;
        sA[i][j] = id - hs * a;
        sC[i][j] = id + hs * a;
        sB[i][j] = id;
    }
    if (tid < SSM_N) {
        Bsh[tid] = Bin[h * SSM_N + tid];
        Cs[tid]  = Cin[h * SSM_N + tid];
    }
    __syncthreads();

    // Gauss-Jordan: sB <- inv(sA). Diagonally dominant -> no pivoting needed.
    for (int k = 0; k < SSM_N; ++k) {
        const float piv = sA[k][k];
        __syncthreads();
        const float rpiv = 1.f / piv;
        if (tid < 64)        sA[k][tid]       *= rpiv;
        else if (tid < 128)  sB[k][tid - 64]  *= rpiv;
        __syncthreads();
        if (tid < 64) fac[tid] = sA[tid][k];
        __syncthreads();
        for (int e = tid; e < 64 * 128; e += 256) {
            const int i = e >> 7, c = e & 127;
            if (i != k) {
                const float f = fac[i];
                if (c < 64) sA[i][c]      -= f * sA[k][c];
                else        sB[i][c - 64] -= f * sB[k][c - 64];
            }
        }
        __syncthreads();
    }

    // Bb = stp * BL @ B
    if (tid < SSM_N) {
        float s = 0.f;
        for (int k2 = 0; k2 < SSM_N; ++k2) s += sB[tid][k2] * Bsh[k2];
        Bbs[tid] = stp * s;
    }
    // Ab = BL @ (I + hs*A)  -> sA   (WMMA; has internal barriers)
    gemm64(&sB[0][0], &sC[0][0], &sA[0][0]);

    // Phase C: X in sC, column j = Ab^j Bb. 63 serial matvecs.
    if (tid < SSM_N) sC[tid][0] = Bbs[tid];
    __syncthreads();
    {
        const int i = tid >> 2, q = tid & 3;
        for (int j = 0; j < SSM_N - 1; ++j) {
            float s = 0.f;
#pragma unroll
            for (int k2 = 0; k2 < 16; ++k2)
                s += sA[i][q * 16 + k2] * sC[q * 16 + k2][j];
            s += __shfl_xor(s, 1);
            s += __shfl_xor(s, 2);
            if (q == 0) sC[i][j + 1] = s;
            __syncthreads();
        }
    }

    // Phase D: MT = Ab^64 via 6 squarings, alternating sA <-> sB; ends in sA.
    {
        float* cur = &sA[0][0];
        float* oth = &sB[0][0];
        for (int s6 = 0; s6 < 6; ++s6) {
            gemm64(cur, cur, oth);
            float* t = cur; cur = oth; oth = t;
        }
        // cur == &sA[0][0]
    }

    // Phase E: per block b, K[h, 64b + j] = C . X[:, j]; then X <- MT @ X.
    {
        float* MT = &sA[0][0];
        float* Xc = &sC[0][0];
        float* Xn = &sB[0][0];
        const int j = tid >> 2, q = tid & 3;
        for (int b = 0; b < SSM_L / SSM_N; ++b) {
            float s = 0.f;
#pragma unroll
            for (int k2 = 0; k2 < 16; ++k2)
                s += Cs[q * 16 + k2] * Xc[(q * 16 + k2) * LDS_LD + j];
            s += __shfl_xor(s, 1);
            s += __shfl_xor(s, 2);
            if (q == 0) Kbuf[h * SSM_L + b * SSM_N + j] = s;
            if (b != SSM_L / SSM_N - 1) {
                gemm64(MT, Xc, Xn);
                float* t = Xc; Xc = Xn; Xn = t;
            }
        }
    }
}

// Direct causal convolution: y[t] = sum_{l<=t} K[l] u[t-l] + D*u[t].
// K and u for one channel (32 KB) staged in LDS; thread owns t = tid + 256m.
__global__ __launch_bounds__(256)
void ssm_kernel_conv(const float* __restrict__ u,     // (H,L)
                     const float* __restrict__ Kbuf,  // (H,L)
                     const float* __restrict__ Dv,    // (H)
                     float* __restrict__ y) {         // (H,L)
    const int h  = blockIdx.x;
    const int t0 = threadIdx.x;
    __shared__ float Ks[SSM_L];
    __shared__ float Us[SSM_L];
    for (int i = t0; i < SSM_L; i += 256) {
        Ks[i] = Kbuf[h * SSM_L + i];
        Us[i] = u[h * SSM_L + i];
    }
    __syncthreads();

    float acc[16];
#pragma unroll
    for (int m = 0; m < 16; ++m) acc[m] = 0.f;

    for (int l = 0; l < SSM_L; ++l) {
        const float kl = Ks[l];                 // LDS broadcast
        const int mstart = (l - t0 + 255) >> 8; // first m with t0+256m >= l (>=0 always)
        for (int m = mstart; m < 16; ++m)
            acc[m] += kl * Us[t0 + (m << 8) - l];
    }

    const float d = Dv[h];
#pragma unroll
    for (int m = 0; m < 16; ++m) {
        const int t = t0 + (m << 8);
        y[h * SSM_L + t] = acc[m] + d * Us[t];
    }
}

extern "C" void kernel_launch(void* const* d_in, const int* in_sizes, int n_in,
                              void* d_out, int out_size, void* d_ws, size_t ws_size,
                              hipStream_t stream) {
    const float* u   = (const float*)d_in[0];  // (H,L)
    const float* A   = (const float*)d_in[1];  // (H,N,N)
    const float* B   = (const float*)d_in[2];  // (H,N,1)
    const float* C   = (const float*)d_in[3];  // (H,1,N)
    const float* D   = (const float*)d_in[4];  // (H,1)
    const float* ls  = (const float*)d_in[5];  // (H,1)
    float* out = (float*)d_out;                // (H,L)
    float* Kbuf = (float*)d_ws;                // (H,L) scratch, 4 MB

    ssm_kernel_K<<<SSM_H, 256, 0, stream>>>(A, B, C, ls, Kbuf);
    ssm_kernel_conv<<<SSM_H, 256, 0, stream>>>(u, Kbuf, D, out);
}